// GateBase_19662360281246
// MI455X (gfx1250) — compile-verified
//
#include <hip/hip_runtime.h>
#include <stdint.h>

typedef __attribute__((ext_vector_type(2))) float v2f;
typedef __attribute__((ext_vector_type(8))) float v8f;

#define NSTATE 16777216   // 4^12 amplitudes per plane
#define DGATE  1024       // 4^5
#define NREST  16384      // 4^7

#define M_BLK 64
#define N_BLK 128
#define KC    16
#define NC    (DGATE / KC)
#define APAD  20          // padded LDS row stride (words): 16B-aligned rows, conflict-free

#define B_COMP_BYTES (N_BLK * APAD * 4)      // 10240: re->im plane delta in sB
#define B_BUF_BYTES  (2 * B_COMP_BYTES)      // 20480: buf0->buf1 delta in sB

// element offset of gate index i (5 base-4 digits -> target axes 0,3,5,8,11)
__device__ __forceinline__ int toff(int i) {
    return ((i >> 8) & 3) * 4194304   // axis 0:  4^11
         + ((i >> 6) & 3) * 65536     // axis 3:  4^8
         + ((i >> 4) & 3) * 4096      // axis 5:  4^6
         + ((i >> 2) & 3) * 64        // axis 8:  4^3
         + (i & 3);                   // axis 11: 4^0
}

__device__ __forceinline__ v8f wmma4(v2f a, v2f b, v8f c) {
    return __builtin_amdgcn_wmma_f32_16x16x4_f32(false, a, false, b, (short)0, c,
                                                 false, false);
}

__global__ __launch_bounds__(256) void gate5_wmma_karatsuba(
    const float* __restrict__ qs_re, const float* __restrict__ qs_im,
    const float* __restrict__ gate_re, const float* __restrict__ gate_im,
    float* __restrict__ out_re, float* __restrict__ out_im)
{
    __shared__ float sA[2][2][M_BLK][APAD];   // [buf][re/im][m][k]
    __shared__ float sB[2][2][N_BLK][APAD];   // [buf][re/im][n][k]
    __shared__ int   sRoff[N_BLK];

    const int tid    = threadIdx.x;
    const int lane   = tid & 31;
    const int wave   = tid >> 5;
    const int lane16 = lane & 31 & 15;
    const int ksel   = (lane >= 16) ? 2 : 0;   // fp32 frag: hi half-wave holds K2,K3
    const int m0     = blockIdx.y * M_BLK;
    const int n0     = blockIdx.x * N_BLK;
    const int m_sub  = (wave & 1) * 32;        // 2 wave rows
    const int n_sub  = (wave >> 1) * 32;       // 4 wave cols

    // rest-axis (N) element offsets: axes 1,2,4,6,7,9,10
    if (tid < N_BLK) {
        const int r = n0 + tid;
        sRoff[tid] = ((r >> 12) & 3) * 1048576
                   + ((r >> 10) & 3) * 262144
                   + ((r >> 8) & 3)  * 16384
                   + ((r >> 6) & 3)  * 1024
                   + ((r >> 4) & 3)  * 256
                   + ((r >> 2) & 3)  * 16
                   + (r & 3)         * 4;
    }
    __syncthreads();

    // ---- per-thread chunk-invariant gather addressing ----
    // addr(k0+kk, n) = toff_hi(k0) [uniform, -> SADDR] + toff_lo(kk) + roff(n)
    const int kk    = tid & 15;
    const int nbase = tid >> 4;                         // 0..15
    const int tlo   = ((kk >> 2) & 3) * 64 + (kk & 3);  // toff_lo
    int      voff[8];      // per-row byte offset into state plane (VGPR, fixed)
    unsigned ldsa0[8];     // LDS dest addr, buffer 0, re-plane
    unsigned ldsa1[8];     // LDS dest addr, buffer 1, re-plane
    #pragma unroll
    for (int j = 0; j < 8; ++j) {
        const int n = nbase + j * 16;                   // 0..127
        voff[j]  = (tlo + sRoff[n]) * 4;
        ldsa0[j] = (unsigned)(uintptr_t)&sB[0][0][n][kk];
        ldsa1[j] = ldsa0[j] + B_BUF_BYTES;
    }

    // async gather of the state tile: 16 bare instructions per chunk,
    // scalar-only per-chunk address update (GVS: SGPR64 + VGPR32 + IOFFSET).
    // INST_OFFSET is added to BOTH lds and global addr -> pre-subtract on im base.
    auto stage_async_B = [&](int cs, int nb) {
        const int k0 = cs * KC;
        const int th = ((k0 >> 8) & 3) * 4194304
                     + ((k0 >> 6) & 3) * 65536
                     + ((k0 >> 4) & 3) * 4096;          // wave-uniform
        const char* base_r = (const char*)qs_re + (size_t)th * 4;
        const char* base_i = (const char*)qs_im + (size_t)th * 4 - B_COMP_BYTES;
        if (nb == 0) {
            #pragma unroll
            for (int j = 0; j < 8; ++j) {
                asm volatile("global_load_async_to_lds_b32 %0, %1, %2"
                             :: "v"(ldsa0[j]), "v"(voff[j]), "s"(base_r) : "memory");
                asm volatile("global_load_async_to_lds_b32 %0, %1, %2 offset:10240"
                             :: "v"(ldsa0[j]), "v"(voff[j]), "s"(base_i) : "memory");
            }
        } else {
            #pragma unroll
            for (int j = 0; j < 8; ++j) {
                asm volatile("global_load_async_to_lds_b32 %0, %1, %2"
                             :: "v"(ldsa1[j]), "v"(voff[j]), "s"(base_r) : "memory");
                asm volatile("global_load_async_to_lds_b32 %0, %1, %2 offset:10240"
                             :: "v"(ldsa1[j]), "v"(voff[j]), "s"(base_i) : "memory");
            }
        }
    };

    // gate staging: 4 consecutive k per thread per component (b128 each way)
    const int gm = (tid * 4) >> 4;   // 0..63
    const int gk = (tid * 4) & 15;   // 0,4,8,12
    float4 g_r, g_i;
    auto load_gate = [&](int k0) {
        g_r = *(const float4*)(gate_re + (size_t)(m0 + gm) * DGATE + k0 + gk);
        g_i = *(const float4*)(gate_im + (size_t)(m0 + gm) * DGATE + k0 + gk);
    };
    auto store_gate = [&](int nb) {
        *(float4*)&sA[nb][0][gm][gk] = g_r;
        *(float4*)&sA[nb][1][gm][gk] = g_i;
    };

    v8f t1[2][2] = {};   // sum Gr*Sr
    v8f t2[2][2] = {};   // sum Gi*Si
    v8f t3[2][2] = {};   // sum (Gr+Gi)*(Sr+Si)

    // prologue: stage chunk 0
    load_gate(0);
    stage_async_B(0, 0);
    store_gate(0);
    asm volatile("s_wait_asynccnt 0x0" ::: "memory");
    __syncthreads();

    for (int c = 0; c < NC; ++c) {
        const int p  = c & 1;
        const int np = p ^ 1;
        if (c + 1 < NC) {                       // prefetch next chunk (overlaps WMMA)
            load_gate((c + 1) * KC);
            stage_async_B(c + 1, np);
        }
        #pragma unroll
        for (int ks = 0; ks < 4; ++ks) {
            const int kb = ks * 4 + ksel;
            v2f ar[2], ai[2], as2[2], br[2], bi[2], bs2[2];
            #pragma unroll
            for (int mt = 0; mt < 2; ++mt) {
                const int row = m_sub + mt * 16 + lane16;
                ar[mt]  = *(const v2f*)&sA[p][0][row][kb];
                ai[mt]  = *(const v2f*)&sA[p][1][row][kb];
                as2[mt] = ar[mt] + ai[mt];      // Karatsuba sum fragment
            }
            #pragma unroll
            for (int nt = 0; nt < 2; ++nt) {
                const int row = n_sub + nt * 16 + lane16;
                br[nt]  = *(const v2f*)&sB[p][0][row][kb];
                bi[nt]  = *(const v2f*)&sB[p][1][row][kb];
                bs2[nt] = br[nt] + bi[nt];
            }
            #pragma unroll
            for (int mt = 0; mt < 2; ++mt)
                #pragma unroll
                for (int nt = 0; nt < 2; ++nt) {
                    t1[mt][nt] = wmma4(ar[mt],  br[nt],  t1[mt][nt]);
                    t2[mt][nt] = wmma4(ai[mt],  bi[nt],  t2[mt][nt]);
                    t3[mt][nt] = wmma4(as2[mt], bs2[nt], t3[mt][nt]);
                }
        }
        if (c + 1 < NC) store_gate(np);
        asm volatile("s_wait_asynccnt 0x0" ::: "memory");
        __syncthreads();
    }

    // epilogue: D layout VGPR v -> M = v (+8 hi half-wave), N = lane16.
    // toff(mb+v) for v=0..3 is contiguous; v=4..7 is contiguous at +64.
    #pragma unroll
    for (int mt = 0; mt < 2; ++mt) {
        const int mb  = m0 + m_sub + mt * 16 + ((lane >= 16) ? 8 : 0);
        const int alb = toff(mb);
        #pragma unroll
        for (int nt = 0; nt < 2; ++nt) {
            const int nref = sRoff[n_sub + nt * 16 + lane16];
            const int alo  = alb + nref;
            float re[8], im[8];
            #pragma unroll
            for (int v = 0; v < 8; ++v) {
                const float a1 = t1[mt][nt][v];
                const float a2 = t2[mt][nt][v];
                const float a3 = t3[mt][nt][v];
                re[v] = a1 - a2;
                im[v] = a3 - a1 - a2;
            }
            *(float4*)(out_re + alo)      = make_float4(re[0], re[1], re[2], re[3]);
            *(float4*)(out_re + alo + 64) = make_float4(re[4], re[5], re[6], re[7]);
            *(float4*)(out_im + alo)      = make_float4(im[0], im[1], im[2], im[3]);
            *(float4*)(out_im + alo + 64) = make_float4(im[4], im[5], im[6], im[7]);
        }
    }
}

extern "C" void kernel_launch(void* const* d_in, const int* in_sizes, int n_in,
                              void* d_out, int out_size, void* d_ws, size_t ws_size,
                              hipStream_t stream) {
    const float* qs_re   = (const float*)d_in[0];
    const float* qs_im   = (const float*)d_in[1];
    const float* gate_re = (const float*)d_in[2];
    const float* gate_im = (const float*)d_in[3];
    float* out_re = (float*)d_out;
    float* out_im = out_re + NSTATE;

    dim3 grid(NREST / N_BLK, DGATE / M_BLK);  // (128, 16)
    gate5_wmma_karatsuba<<<grid, 256, 0, stream>>>(qs_re, qs_im, gate_re, gate_im,
                                                   out_re, out_im);
}